// DW5_51_68186900792258
// MI455X (gfx1250) — compile-verified
//
#include <hip/hip_runtime.h>

// Depthwise 5x5 + 51x51 fp32 conv via V_WMMA_F32_16X16X4_F32.
// One block = one (image,channel) HALF plane (64 wide x 32 tall).
// 128 threads = 4 waves; wave w owns tiles (ylocal 0..15) and (16..31) at x0=16w.
// Banded Toeplitz weights stored zero-padded in LDS -> branch-free B loads.
// Input plane staged by the Tensor Data Mover (pad_enable matches LDS stride).

#define CCH 192
#define HW  64
#define LSTRIDE 115              // 114 live cols + 1 pad col
#define NROWS   82               // 32 + 50 haloed rows
#define K51 51
#define K5  5

#define OFF_W5   0               // 5 rows * stride 20 + 15 tail reach -> 115, pad to 128
#define OFF_W51  128             // 51 rows * stride 68 + head pad 17 -> 3485, pad to 3488
#define W51LEN   3488
#define OFF_IN   (OFF_W51 + W51LEN)          // 3616
#define SMEM_TOT (OFF_IN + NROWS * LSTRIDE)  // 3616 + 9430 = 13046 floats = 52184 B

typedef __attribute__((ext_vector_type(2))) float v2f;
typedef __attribute__((ext_vector_type(8))) float v8f;
typedef __attribute__((ext_vector_type(4))) unsigned int v4u;
typedef __attribute__((ext_vector_type(4))) int v4i;
typedef __attribute__((ext_vector_type(8))) int v8i;

__global__ __launch_bounds__(128)
void dw5_51_wmma_f32(const float* __restrict__ x,
                     const float* __restrict__ w5,
                     const float* __restrict__ w51,
                     float* __restrict__ out) {
  __shared__ float smem[SMEM_TOT];

  const int blk  = blockIdx.x;
  const int half = blk & 1;
  const int bc   = blk >> 1;                 // (img*C + ch)
  const int ch   = bc % CCH;
  const int tid  = threadIdx.x;
  const int y0h  = half * 32;

  const float* __restrict__ xin = x + (size_t)bc * (HW * HW);

  // ---- phase 1: zero all LDS (halo + weight pads must be 0) ----
  for (int i = tid; i < SMEM_TOT; i += 128) smem[i] = 0.0f;
  __syncthreads();

  // ---- phase 2: scatter zero-padded Toeplitz weights; TDM stages the plane ----
  for (int i = tid; i < K51 * K51; i += 128) {
    int r = i / K51, c = i - r * K51;
    smem[OFF_W51 + r * 68 + 17 + c] = w51[ch * (K51 * K51) + i];
  }
  if (tid < K5 * K5) {
    int r = tid / K5, c = tid - r * K5;
    smem[OFF_W5 + r * 20 + 15 + c] = w5[ch * (K5 * K5) + tid];
  }

  if (tid < 32) {   // wave 0 issues the tensor DMA (EXEC-independent, wave-level)
    int gy0 = (y0h - 25 < 0) ? 0 : (y0h - 25);      // first live global row
    int gy1 = (y0h + 56 > 63) ? 63 : (y0h + 56);    // last live global row
    unsigned nrows   = (unsigned)(gy1 - gy0 + 1);   // 57
    unsigned ldsrow0 = (unsigned)(gy0 - (y0h - 25));
    unsigned lds_addr = (unsigned)((OFF_IN + ldsrow0 * LSTRIDE + 25) * 4);
    unsigned long long ga = (unsigned long long)(const void*)(xin + gy0 * HW);

    // D# group 0: count=1 | lds_addr | global_addr | type=2
    v4u g0;
    g0.x = 1u;
    g0.y = lds_addr;
    g0.z = (unsigned)(ga & 0xffffffffu);
    g0.w = (unsigned)((ga >> 32) & 0x01ffffffu) | 0x80000000u;

    // D# group 1: data_size=4B, pad_enable, pad_interval=64 dw, pad_amount=51 dw,
    //             tensor 64 x nrows, tile 64 x nrows, dim0 stride 64
    v8i g1;
    g1[0] = (int)((2u << 16) | (1u << 20) | (5u << 22) | (50u << 25));
    g1[1] = (int)(64u << 16);                 // tensor_dim0[15:0] @ bits 63:48
    g1[2] = (int)((nrows & 0xffffu) << 16);   // tensor_dim1[15:0] @ bits 95:80
    g1[3] = (int)(64u << 16);                 // tensor_dim1 hi=0, tile_dim0=64 @127:112
    g1[4] = (int)(nrows & 0xffffu);           // tile_dim1 @143:128, tile_dim2=0
    g1[5] = 64;                               // tensor_dim0_stride[31:0]
    g1[6] = 0;
    g1[7] = 0;

    v4i g2 = {0, 0, 0, 0};                    // 2D: groups 2/3 unused
    v4i g3 = {0, 0, 0, 0};
    v8i g4 = {0, 0, 0, 0, 0, 0, 0, 0};        // extra group of 6-arg form: unused
    __builtin_amdgcn_tensor_load_to_lds(g0, g1, g2, g3, g4, 0);
    __builtin_amdgcn_s_wait_tensorcnt(0);
  }
  __syncthreads();

  // ---- per-wave tiles ----
  const int lane = tid & 31;
  const int n = lane & 15;      // N col of B/C/D == M row of A operand
  const int g = lane >> 4;      // lane half -> K pair {2g, 2g+1}
  const int x0 = (tid >> 5) * 16;

  v8f acc0 = {0.f,0.f,0.f,0.f,0.f,0.f,0.f,0.f};
  v8f acc1 = {0.f,0.f,0.f,0.f,0.f,0.f,0.f,0.f};

  // ================= 51x51 tap: kk window shifted by -2 =================
  // A col = x0 + kk - 2 ; B = wpad[dy*68 + 15 + kk - n] (zero outside band)
  {
    int bbase = OFF_W51 + 15 + 2 * g - n;
    int abase = OFF_IN + n * LSTRIDE + x0 - 2 + 2 * g;
    for (int dy = 0; dy < K51; ++dy) {
      const float* __restrict__ wrow = &smem[bbase];
      const float* __restrict__ ar0  = &smem[abase];
      const float* __restrict__ ar1  = ar0 + 16 * LSTRIDE;
#pragma unroll
      for (int kt = 0; kt < 17; ++kt) {
        v2f b;  b.x  = wrow[4 * kt]; b.y  = wrow[4 * kt + 1];
        v2f a0; a0.x = ar0[4 * kt];  a0.y = ar0[4 * kt + 1];
        acc0 = __builtin_amdgcn_wmma_f32_16x16x4_f32(
            false, a0, false, b, (short)0, acc0, false, false);
        v2f a1; a1.x = ar1[4 * kt];  a1.y = ar1[4 * kt + 1];
        acc1 = __builtin_amdgcn_wmma_f32_16x16x4_f32(
            false, a1, false, b, (short)0, acc1, false, false);
      }
      bbase += 68;
      abase += LSTRIDE;
    }
  }

  // ================= 5x5 tap (halo offset 25-2 = 23) =================
  {
    int bbase = OFF_W5 + 15 + 2 * g - n;
    int abase = OFF_IN + (n + 23) * LSTRIDE + x0 + 23 + 2 * g;
    for (int dy = 0; dy < K5; ++dy) {
      const float* __restrict__ wrow = &smem[bbase];
      const float* __restrict__ ar0  = &smem[abase];
      const float* __restrict__ ar1  = ar0 + 16 * LSTRIDE;
#pragma unroll
      for (int kt = 0; kt < 5; ++kt) {
        v2f b;  b.x  = wrow[4 * kt]; b.y  = wrow[4 * kt + 1];
        v2f a0; a0.x = ar0[4 * kt];  a0.y = ar0[4 * kt + 1];
        acc0 = __builtin_amdgcn_wmma_f32_16x16x4_f32(
            false, a0, false, b, (short)0, acc0, false, false);
        v2f a1; a1.x = ar1[4 * kt];  a1.y = ar1[4 * kt + 1];
        acc1 = __builtin_amdgcn_wmma_f32_16x16x4_f32(
            false, a1, false, b, (short)0, acc1, false, false);
      }
      bbase += 20;
      abase += LSTRIDE;
    }
  }

  // ---- store D: VGPR v holds rows v (lanes 0-15) / v+8 (lanes 16-31) ----
  float* __restrict__ outp = out + (size_t)bc * (HW * HW) + y0h * HW;
#pragma unroll
  for (int v = 0; v < 8; ++v) {
    outp[(v + 8 * g) * HW + x0 + n]      = acc0[v];
    outp[(16 + v + 8 * g) * HW + x0 + n] = acc1[v];
  }
}

extern "C" void kernel_launch(void* const* d_in, const int* in_sizes, int n_in,
                              void* d_out, int out_size, void* d_ws, size_t ws_size,
                              hipStream_t stream) {
  const float* x   = (const float*)d_in[0];
  const float* w5  = (const float*)d_in[1];
  const float* w51 = (const float*)d_in[2];
  float* out = (float*)d_out;

  dim3 grid(32 * CCH * 2);   // one block per (image, channel, half-plane)
  dim3 block(128);
  dw5_51_wmma_f32<<<grid, block, 0, stream>>>(x, w5, w51, out);
}